// ESN_70703751627213
// MI455X (gfx1250) — compile-verified
//
#include <hip/hip_runtime.h>
#include <math.h>

// ESN fused kernel for MI455X (gfx1250, wave32).
// B=32, D_IN=64, D_STATE=1024, L=2048, leak=0.5.
// out[b,p,l]: x <- 0.5 x + 0.5 tanh(w_in[p,:]·u[b,:,l] + diag(w_hh)[p]*x + bias[p])

#define B_SZ     32
#define D_IN     64
#define D_ST     1024
#define L_SEQ    2048
#define LEAK     0.5f

#define P_TILE   256    // states per workgroup
#define L_TILE   32     // timesteps per chunk
#define NTHREADS 256    // 8 wave32
#define KP       32     // D_IN/2 K-pairs
#define UP_STRIDE 96    // dwords per kp row (64 used + 32 pad): rows kp,kp+1 map to disjoint bank halves
#define X_STRIDE 34     // even padded stride: per-wave conflict-free AND 8B-aligned timestep pairs
#define U_BUF    (KP * UP_STRIDE)

typedef float v2f __attribute__((ext_vector_type(2)));
typedef float v8f __attribute__((ext_vector_type(8)));

#if __has_builtin(__builtin_amdgcn_tanhf)
#define TANH(x) __builtin_amdgcn_tanhf(x)
#else
#define TANH(x) tanhf(x)
#endif

#if __has_builtin(__builtin_amdgcn_global_load_async_to_lds_b32) && \
    __has_builtin(__builtin_amdgcn_s_wait_asynccnt)
#define HAVE_ASYNC 1
#else
#define HAVE_ASYNC 0
#endif

#if HAVE_ASYNC
typedef __attribute__((address_space(1))) int GI;   // global (AS1) int, per builtin signature
typedef __attribute__((address_space(3))) int LI;   // LDS (AS3) int
// Issue the interleaved u-tile copy for one chunk: u rows 2kp and 2kp+1 go column-interleaved into LDS row kp.
// Per-lane LDS scatter is legal for async-to-LDS (VDST is a per-lane LDS address).
__device__ __forceinline__ void issue_u_async(const float* ub, int l0, float* dst, int t) {
#pragma unroll
    for (int rep = 0; rep < 8; ++rep) {
        int i   = t + rep * NTHREADS;   // 0..2047
        int h   = i >> 5;               // 0..63 (global row; wave reads 128B contiguous)
        int col = i & 31;
        const float* g = ub + (size_t)h * L_SEQ + l0 + col;
        float*       l = dst + (h >> 1) * UP_STRIDE + 2 * col + (h & 1);
        __builtin_amdgcn_global_load_async_to_lds_b32((GI*)g, (LI*)l, 0, 0);
    }
}
#endif

__global__ __launch_bounds__(NTHREADS) void esn_fused_kernel(
    const float* __restrict__ u,     // [B, D_IN, L]
    const float* __restrict__ w_in,  // [D_STATE, D_IN]
    const float* __restrict__ w_hh,  // [D_STATE, D_STATE] (diagonal used)
    const float* __restrict__ bias,  // [D_STATE]
    float* __restrict__ out)         // [B, D_STATE, L]
{
    __shared__ float x_lds[P_TILE * X_STRIDE];   // 34.8 KB (u_proj, overwritten in-place by x)
#if HAVE_ASYNC
    __shared__ float u_lds[2 * U_BUF];           // 24 KB double-buffered u tile
#else
    __shared__ float u_lds[U_BUF];               // 12 KB
#endif

    const int t     = threadIdx.x;
    const int wave  = t >> 5;
    const int lane  = t & 31;
    const int laneM = lane & 15;   // M (A) / N (B,C) index within a 16-wide tile
    const int khalf = lane >> 4;   // which K half the upper 16 lanes carry

    const int b  = blockIdx.y;
    const int p0 = blockIdx.x * P_TILE;

    // ---- Preload A fragments: this wave's 32 w_in rows, two 16-row M tiles, K=64 in 16 steps of 4.
    // fp32 A 16x4 layout: lanes 0-15 hold (K=0,K=1), lanes 16-31 hold (K=2,K=3).
    v2f a[2][16];
#pragma unroll
    for (int mt = 0; mt < 2; ++mt) {
        const float* wrow = w_in + (size_t)(p0 + wave * 32 + mt * 16 + laneM) * D_IN;
#pragma unroll
        for (int k = 0; k < 16; ++k)
            a[mt][k] = *(const v2f*)(wrow + k * 4 + khalf * 2);
    }

    // ---- Per-state recurrence constants (thread t owns state p).
    const int   p  = p0 + t;
    const float d  = w_hh[(size_t)p * D_ST + p];   // diagonal element
    const float bi = bias[p];
    float x = 0.0f;

    const float* ub = u + (size_t)b * D_IN * L_SEQ;
    float*       ob = out + ((size_t)b * D_ST + p0) * L_SEQ;

#if HAVE_ASYNC
    issue_u_async(ub, 0, &u_lds[0], t);   // prefetch chunk 0
    int cur = 0;
#endif

    for (int l0 = 0; l0 < L_SEQ; l0 += L_TILE) {
#if HAVE_ASYNC
        __builtin_amdgcn_s_wait_asynccnt(0);
        float* ubuf = &u_lds[cur * U_BUF];
#else
        // Synchronous staging, K-pair interleaved: u_lds[kp][2c+j] = u[2kp+j][l0+c].
#pragma unroll
        for (int rep = 0; rep < 2; ++rep) {
            int i  = t + rep * NTHREADS;   // 0..511
            int kp = i >> 4;               // 0..31
            int c2 = (i & 15) * 2;         // 0..30
            v2f ga = *(const v2f*)(ub + (size_t)(2 * kp)     * L_SEQ + l0 + c2);
            v2f gb = *(const v2f*)(ub + (size_t)(2 * kp + 1) * L_SEQ + l0 + c2);
            float4 v; v.x = ga.x; v.y = gb.x; v.z = ga.y; v.w = gb.y;
            *(float4*)(&u_lds[kp * UP_STRIDE + 2 * c2]) = v;
        }
        float* ubuf = u_lds;
#endif
        __syncthreads();   // u tile ready; also fences prev iter's x_lds readers vs. new stores

        // ---- WMMA: up[256x32] = w_in_tile[256x64] * u_tile[64x32], fp32.
        // B fragments 1-deep software-pipelined so ds_load overlaps the WMMA pipe.
        v8f acc[2][2];
#pragma unroll
        for (int mt = 0; mt < 2; ++mt)
#pragma unroll
            for (int nt = 0; nt < 2; ++nt) {
                v8f z = {0.f, 0.f, 0.f, 0.f, 0.f, 0.f, 0.f, 0.f};
                acc[mt][nt] = z;
            }

        v2f bfr[2][2];
        bfr[0][0] = *(const v2f*)(ubuf + khalf * UP_STRIDE + 2 * laneM);
        bfr[0][1] = *(const v2f*)(ubuf + khalf * UP_STRIDE + 2 * (16 + laneM));
#pragma unroll
        for (int k = 0; k < 16; ++k) {
            const int cb = k & 1;
            if (k < 15) {
                const int kpn = (k + 1) * 2 + khalf;
                bfr[cb ^ 1][0] = *(const v2f*)(ubuf + kpn * UP_STRIDE + 2 * laneM);
                bfr[cb ^ 1][1] = *(const v2f*)(ubuf + kpn * UP_STRIDE + 2 * (16 + laneM));
            }
#pragma unroll
            for (int mt = 0; mt < 2; ++mt)
#pragma unroll
                for (int nt = 0; nt < 2; ++nt)
                    acc[mt][nt] = __builtin_amdgcn_wmma_f32_16x16x4_f32(
                        false, a[mt][k], false, bfr[cb][nt],
                        (short)0, acc[mt][nt], false, false);
        }

        // ---- Scatter accumulators to LDS in [p][l] layout.
        // f32 C/D layout: VGPR r holds row M=r (lanes 0-15) / M=r+8 (lanes 16-31), N = lane%16.
#pragma unroll
        for (int mt = 0; mt < 2; ++mt)
#pragma unroll
            for (int nt = 0; nt < 2; ++nt)
#pragma unroll
                for (int r = 0; r < 8; ++r) {
                    int row = wave * 32 + mt * 16 + khalf * 8 + r;
                    x_lds[row * X_STRIDE + nt * 16 + laneM] = acc[mt][nt][r];
                }

#if HAVE_ASYNC
        // Prefetch next chunk into the other buffer; overlaps scan + write-out.
        if (l0 + L_TILE < L_SEQ)
            issue_u_async(ub, l0 + L_TILE, &u_lds[(cur ^ 1) * U_BUF], t);
        cur ^= 1;
#endif
        __syncthreads();   // projection visible to scan threads

        // ---- Sequential elementwise recurrence, two timesteps per b64 LDS round trip.
        float* xr = &x_lds[t * X_STRIDE];   // 8B aligned (X_STRIDE even)
#pragma unroll
        for (int l = 0; l < L_TILE; l += 2) {
            v2f pr = *(const v2f*)(xr + l);
            float pre0 = pr.x + fmaf(x, d, bi);
            x = fmaf(LEAK, TANH(pre0), (1.0f - LEAK) * x);
            v2f xo; xo.x = x;
            float pre1 = pr.y + fmaf(x, d, bi);
            x = fmaf(LEAK, TANH(pre1), (1.0f - LEAK) * x);
            xo.y = x;
            *(v2f*)(xr + l) = xo;
        }
        __syncthreads();   // states visible to write-out threads

        // ---- Coalesced write-out: 2048 float4, 8 per thread; 8 threads cover one state's 32 floats.
#pragma unroll
        for (int rep = 0; rep < 8; ++rep) {
            int i   = t + rep * NTHREADS;   // 0..2047
            int row = i >> 3;               // 0..255
            int l4  = (i & 7) * 4;
            v2f lo = *(const v2f*)(&x_lds[row * X_STRIDE + l4]);
            v2f hi = *(const v2f*)(&x_lds[row * X_STRIDE + l4 + 2]);
            float4 v; v.x = lo.x; v.y = lo.y; v.z = hi.x; v.w = hi.y;
            *(float4*)(ob + (size_t)row * L_SEQ + l0 + l4) = v;
        }
        // next iteration's x_lds stores are fenced by the loop-top barrier
    }
}

extern "C" void kernel_launch(void* const* d_in, const int* in_sizes, int n_in,
                              void* d_out, int out_size, void* d_ws, size_t ws_size,
                              hipStream_t stream) {
    const float* u    = (const float*)d_in[0];   // [32, 64, 2048]
    const float* w_in = (const float*)d_in[1];   // [1024, 64]
    const float* w_hh = (const float*)d_in[2];   // [1024, 1024]
    const float* bias = (const float*)d_in[3];   // [1024]
    float* out = (float*)d_out;                  // [32, 1024, 2048]

    dim3 grid(D_ST / P_TILE, B_SZ);  // (4, 32)
    dim3 block(NTHREADS);
    esn_fused_kernel<<<grid, block, 0, stream>>>(u, w_in, w_hh, bias, out);
}